// GraphNet_EQ_18296560681269
// MI455X (gfx1250) — compile-verified
//
#include <hip/hip_runtime.h>
#include <math.h>

#define N_NODES 50000
#define N_EDGES 400000
#define MUL0 64
#define MUL1 32
#define NEF 12
#define RN 100
#define WNUM 192
#define NODE_IN_LD 160   // MUL0 + 3*MUL1
#define H1_LD 96         // (32,3) flattened
#define NS_LD 96
#define NV_LD 288        // (96,3) flattened
#define OUT_LD 160

typedef __attribute__((ext_vector_type(2))) float v2f;
typedef __attribute__((ext_vector_type(8))) float v8f;

// ---- WMMA helpers (V_WMMA_F32_16X16X4_F32, wave32) ----------------------
__device__ __forceinline__ v8f wmma4(v2f a, v2f b, v8f c) {
  // 8 args: (neg_a, A, neg_b, B, c_mod, C, reuse_a, reuse_b)
  return __builtin_amdgcn_wmma_f32_16x16x4_f32(false, a, false, b, (short)0, c,
                                               false, false);
}

// A 16x4 f32 fragment: lanes 0-15 hold row M=lane, K = k0,k0+1;
// lanes 16-31 hold row M=lane-16, K = k0+2,k0+3.
// Source is row-major with element strides (rs per row, ks per K step).
__device__ __forceinline__ v2f load_a_strided(const float* __restrict__ base,
                                              int row0, int rs, int k0, int ks,
                                              int lane) {
  int r = row0 + (lane & 15);
  int k = k0 + ((lane >> 4) << 1);
  const float* p = base + (size_t)r * rs + (size_t)k * ks;
  v2f a;
  a.x = p[0];
  a.y = p[ks];
  return a;
}

// B 4x16 f32 fragment from row-major W (K x Nmat), ld = Nmat.
__device__ __forceinline__ v2f load_b_rm(const float* __restrict__ W, int k0,
                                         int n0, int ld, int lane) {
  int n = n0 + (lane & 15);
  int k = k0 + ((lane >> 4) << 1);
  v2f b;
  b.x = W[(size_t)k * ld + n];
  b.y = W[(size_t)(k + 1) * ld + n];
  return b;
}

// ---- zero the scatter accumulators --------------------------------------
__global__ void zero_kernel(float* __restrict__ p, size_t n) {
  size_t i = (size_t)blockIdx.x * blockDim.x + threadIdx.x;
  size_t stride = (size_t)gridDim.x * blockDim.x;
  for (; i < n; i += stride) p[i] = 0.0f;
}

// ---- node hidden features: h0 = x0@W_l1_0 * z/8 ; h1c = x1c@W_l1_1 * z/sqrt32
__global__ void __launch_bounds__(256) node_l1_kernel(
    const float* __restrict__ node_input, const float* __restrict__ node_attr,
    const float* __restrict__ W_l1_0, const float* __restrict__ W_l1_1,
    float* __restrict__ h0, float* __restrict__ h1) {
  int wave = blockIdx.x * (blockDim.x >> 5) + (threadIdx.x >> 5);
  int lane = threadIdx.x & 31;
  const int ntiles = N_NODES / 16;
  if (wave >= ntiles) return;  // uniform within wave
  int row0 = wave * 16;

  // scalar channel: 16x64 = 4 col tiles, K=64
  for (int ct = 0; ct < 4; ++ct) {
    v8f acc = {};
    for (int ks = 0; ks < MUL0; ks += 4) {
      v2f a = load_a_strided(node_input, row0, NODE_IN_LD, ks, 1, lane);
      v2f b = load_b_rm(W_l1_0, ks, ct * 16, MUL0, lane);
      acc = wmma4(a, b, acc);
    }
    int col = ct * 16 + (lane & 15);
#pragma unroll
    for (int j = 0; j < 8; ++j) {
      int row = row0 + j + ((lane >> 4) << 3);
      float zz = node_attr[row];
      h0[(size_t)row * MUL0 + col] = acc[j] * zz * 0.125f;
    }
  }
  // vector channels: per channel 16x32, K=32
  for (int ch = 0; ch < 3; ++ch) {
    for (int ct = 0; ct < 2; ++ct) {
      v8f acc = {};
      for (int ks = 0; ks < MUL1; ks += 4) {
        v2f a = load_a_strided(node_input + MUL0 + ch, row0, NODE_IN_LD, ks, 3,
                               lane);
        v2f b = load_b_rm(W_l1_1, ks, ct * 16, MUL1, lane);
        acc = wmma4(a, b, acc);
      }
      int u = ct * 16 + (lane & 15);
#pragma unroll
      for (int j = 0; j < 8; ++j) {
        int row = row0 + j + ((lane >> 4) << 3);
        float zz = node_attr[row];
        h1[(size_t)row * H1_LD + u * 3 + ch] =
            acc[j] * zz * 0.17677669529663687f;  // 1/sqrt(32)
      }
    }
  }
}

// ---- fused edge MLP + tensor product + scatter --------------------------
#define WAVES_PER_BLOCK 8
#define H_LDS_LD 104

__global__ void __launch_bounds__(256) edge_kernel(
    const float* __restrict__ edge_features, const float* __restrict__ edge_attr,
    const int* __restrict__ edge_src, const int* __restrict__ edge_dst,
    const float* __restrict__ W_fc1, const float* __restrict__ W_fc2,
    const float* __restrict__ h0, const float* __restrict__ h1,
    float* __restrict__ ns, float* __restrict__ nv) {
  __shared__ float lds_h[WAVES_PER_BLOCK][16][H_LDS_LD];
  __shared__ int lds_src[WAVES_PER_BLOCK][16];
  __shared__ int lds_dst[WAVES_PER_BLOCK][16];
  __shared__ float lds_y[WAVES_PER_BLOCK][16][4];

  int wid = threadIdx.x >> 5;
  int lane = threadIdx.x & 31;
  int ebase = (blockIdx.x * WAVES_PER_BLOCK + wid) * 16;  // E % 128 == 0

  if (lane < 16) {
    int e = ebase + lane;
    lds_src[wid][lane] = edge_src[e];
    lds_dst[wid][lane] = edge_dst[e];
#pragma unroll
    for (int i = 0; i < 4; ++i)
      lds_y[wid][lane][i] = edge_attr[(size_t)e * 4 + i];
  }

  // GEMM1: h = silu(ef @ W_fc1 / sqrt(12)), 16x100, K=12 ; stage in LDS
  for (int ct = 0; ct < 7; ++ct) {
    v8f acc = {};
#pragma unroll
    for (int ks = 0; ks < NEF; ks += 4) {
      v2f a = load_a_strided(edge_features, ebase, NEF, ks, 1, lane);
      int n = ct * 16 + (lane & 15);
      int k = ks + ((lane >> 4) << 1);
      v2f b;
      b.x = (n < RN) ? W_fc1[(size_t)k * RN + n] : 0.0f;
      b.y = (n < RN) ? W_fc1[(size_t)(k + 1) * RN + n] : 0.0f;
      acc = wmma4(a, b, acc);
    }
    int col = ct * 16 + (lane & 15);
    if (col < RN) {
#pragma unroll
      for (int j = 0; j < 8; ++j) {
        int er = j + ((lane >> 4) << 3);
        float v = acc[j] * 0.28867513459481287f;  // 1/sqrt(12)
        lds_h[wid][er][col] = v / (1.0f + __expf(-v));  // silu
      }
    }
  }
  __syncthreads();

  // GEMM2 col-tile by col-tile: w = h @ W_fc2 / sqrt(100); scatter directly
  // from D fragments (lane = output column, regs j = edge rows).
  const float inv_nb = 0.35355339059327373f;  // 1/sqrt(8)
  for (int ct = 0; ct < 12; ++ct) {
    v8f acc = {};
    for (int ks = 0; ks < RN; ks += 4) {
      int k = ks + ((lane >> 4) << 1);
      v2f a;
      a.x = lds_h[wid][lane & 15][k];
      a.y = lds_h[wid][lane & 15][k + 1];
      v2f b = load_b_rm(W_fc2, ks, ct * 16, WNUM, lane);
      acc = wmma4(a, b, acc);
    }
    int colg = ct * 16 + (lane & 15);
#pragma unroll
    for (int j = 0; j < 8; ++j) {
      int er = j + ((lane >> 4) << 3);
      float wv = acc[j] * 0.1f;  // 1/sqrt(100)
      int s = lds_src[wid][er];
      int d = lds_dst[wid][er];
      float y0 = lds_y[wid][er][0];
      float y1a = lds_y[wid][er][1];
      float y1b = lds_y[wid][er][2];
      float y1c = lds_y[wid][er][3];
      if (ct < 4) {  // w0: m0a -> ns[:, 0:64]
        int c = colg;
        float e0 = h0[(size_t)s * MUL0 + c];
        unsafeAtomicAdd(&ns[(size_t)d * NS_LD + c], wv * e0 * y0 * inv_nb);
      } else if (ct < 8) {  // w1: m1a -> nv[:, 0:64, :]
        int c = colg - 64;
        float e0 = h0[(size_t)s * MUL0 + c];
        float t = wv * e0 * inv_nb;
        float* base = &nv[(size_t)d * NV_LD + c * 3];
        unsafeAtomicAdd(base + 0, t * y1a);
        unsafeAtomicAdd(base + 1, t * y1b);
        unsafeAtomicAdd(base + 2, t * y1c);
      } else if (ct < 10) {  // w2: m1b -> nv[:, 64:96, :]
        int u = colg - 128;
        const float* e1p = &h1[(size_t)s * H1_LD + u * 3];
        float t = wv * y0 * inv_nb;
        float* base = &nv[(size_t)d * NV_LD + (64 + u) * 3];
        unsafeAtomicAdd(base + 0, t * e1p[0]);
        unsafeAtomicAdd(base + 1, t * e1p[1]);
        unsafeAtomicAdd(base + 2, t * e1p[2]);
      } else {  // w3: m0b -> ns[:, 64:96]
        int u = colg - 160;
        const float* e1p = &h1[(size_t)s * H1_LD + u * 3];
        float dot = e1p[0] * y1a + e1p[1] * y1b + e1p[2] * y1c;
        unsafeAtomicAdd(&ns[(size_t)d * NS_LD + 64 + u],
                        wv * dot * 0.5773502691896258f * inv_nb);  // 1/sqrt(3)
      }
    }
  }
}

// ---- output: out = z*(cs*sc + cx*l2) ------------------------------------
__global__ void __launch_bounds__(256) node_out_kernel(
    const float* __restrict__ node_input, const float* __restrict__ node_attr,
    const float* __restrict__ W_sc0, const float* __restrict__ W_sc1,
    const float* __restrict__ W_l2_0, const float* __restrict__ W_l2_1,
    const float* __restrict__ ns, const float* __restrict__ nv,
    float* __restrict__ out) {
  int wave = blockIdx.x * (blockDim.x >> 5) + (threadIdx.x >> 5);
  int lane = threadIdx.x & 31;
  const int ntiles = N_NODES / 16;
  if (wave >= ntiles) return;
  int row0 = wave * 16;
  const float k_s0 = 0.3826834323650898f * 0.125f;                // cs/sqrt(64)
  const float k_o = 0.9238795325112867f * 0.10206207261596575f;   // cx/sqrt(96)
  const float k_s1 = 0.3826834323650898f * 0.17677669529663687f;  // cs/sqrt(32)

  // scalar: out[:, :64]
  for (int ct = 0; ct < 4; ++ct) {
    v8f accS = {};
    v8f accO = {};
    for (int ks = 0; ks < MUL0; ks += 4) {
      v2f a = load_a_strided(node_input, row0, NODE_IN_LD, ks, 1, lane);
      v2f b = load_b_rm(W_sc0, ks, ct * 16, MUL0, lane);
      accS = wmma4(a, b, accS);
    }
    for (int ks = 0; ks < 96; ks += 4) {
      v2f a = load_a_strided(ns, row0, NS_LD, ks, 1, lane);
      v2f b = load_b_rm(W_l2_0, ks, ct * 16, MUL0, lane);
      accO = wmma4(a, b, accO);
    }
    int col = ct * 16 + (lane & 15);
#pragma unroll
    for (int j = 0; j < 8; ++j) {
      int row = row0 + j + ((lane >> 4) << 3);
      float zz = node_attr[row];
      out[(size_t)row * OUT_LD + col] = zz * (k_s0 * accS[j] + k_o * accO[j]);
    }
  }
  // vector: out[:, 64:160] interleaved (u,3)
  for (int ch = 0; ch < 3; ++ch) {
    for (int ct = 0; ct < 2; ++ct) {
      v8f accS = {};
      v8f accO = {};
      for (int ks = 0; ks < MUL1; ks += 4) {
        v2f a = load_a_strided(node_input + MUL0 + ch, row0, NODE_IN_LD, ks, 3,
                               lane);
        v2f b = load_b_rm(W_sc1, ks, ct * 16, MUL1, lane);
        accS = wmma4(a, b, accS);
      }
      for (int ks = 0; ks < 96; ks += 4) {
        v2f a = load_a_strided(nv + ch, row0, NV_LD, ks, 3, lane);
        v2f b = load_b_rm(W_l2_1, ks, ct * 16, MUL1, lane);
        accO = wmma4(a, b, accO);
      }
      int u = ct * 16 + (lane & 15);
#pragma unroll
      for (int j = 0; j < 8; ++j) {
        int row = row0 + j + ((lane >> 4) << 3);
        float zz = node_attr[row];
        out[(size_t)row * OUT_LD + MUL0 + u * 3 + ch] =
            zz * (k_s1 * accS[j] + k_o * accO[j]);
      }
    }
  }
}

extern "C" void kernel_launch(void* const* d_in, const int* in_sizes, int n_in,
                              void* d_out, int out_size, void* d_ws,
                              size_t ws_size, hipStream_t stream) {
  const float* node_input = (const float*)d_in[0];
  const float* node_attr = (const float*)d_in[1];
  const int* edge_src = (const int*)d_in[2];
  const int* edge_dst = (const int*)d_in[3];
  const float* edge_attr = (const float*)d_in[4];
  const float* edge_feat = (const float*)d_in[5];
  const float* W_sc0 = (const float*)d_in[6];
  const float* W_sc1 = (const float*)d_in[7];
  const float* W_l1_0 = (const float*)d_in[8];
  const float* W_l1_1 = (const float*)d_in[9];
  const float* W_fc1 = (const float*)d_in[10];
  const float* W_fc2 = (const float*)d_in[11];
  const float* W_l2_0 = (const float*)d_in[12];
  const float* W_l2_1 = (const float*)d_in[13];
  float* out = (float*)d_out;

  float* ws = (float*)d_ws;
  float* h0 = ws;                                  // N*64
  float* h1 = h0 + (size_t)N_NODES * MUL0;         // N*96
  float* ns = h1 + (size_t)N_NODES * H1_LD;        // N*96
  float* nv = ns + (size_t)N_NODES * NS_LD;        // N*288
  size_t acc_count = (size_t)N_NODES * (NS_LD + NV_LD);

  const int node_tiles = N_NODES / 16;                      // 3125
  const int node_blocks = (node_tiles + 7) / 8;             // 8 waves/block
  const int edge_blocks = N_EDGES / (WAVES_PER_BLOCK * 16); // 3125, exact

  zero_kernel<<<2048, 256, 0, stream>>>(ns, acc_count);
  node_l1_kernel<<<node_blocks, 256, 0, stream>>>(node_input, node_attr,
                                                  W_l1_0, W_l1_1, h0, h1);
  edge_kernel<<<edge_blocks, 256, 0, stream>>>(edge_feat, edge_attr, edge_src,
                                               edge_dst, W_fc1, W_fc2, h0, h1,
                                               ns, nv);
  node_out_kernel<<<node_blocks, 256, 0, stream>>>(node_input, node_attr,
                                                   W_sc0, W_sc1, W_l2_0,
                                                   W_l2_1, ns, nv, out);
}